// WeightsRegLoss_12335146074572
// MI455X (gfx1250) — compile-verified
//
#include <hip/hip_runtime.h>

// Problem constants from the reference
#define IMG_H 960
#define IMG_W 540
#define NCH   192
#define WIN   20

constexpr int CELLS_1D = WIN - 1;                 // 19
constexpr int CHUNKS   = NCH / 4;                 // 48 float4 chunks per pixel
constexpr int ROWTASKS = CELLS_1D * CHUNKS;       // 912 (j, chunk) tasks per row
constexpr int BLOCK    = 1024;                    // 32 wave32s on one WGP

typedef __attribute__((ext_vector_type(2))) float v2f;
typedef __attribute__((ext_vector_type(8))) float v8f;

// Full 32-lane wave sum using V_WMMA_F32_16X16X4_F32 as the reduction engine.
//   A[m,0] = x(lane m), A[m,2] = x(lane m+16), A[*,1]=A[*,3]=0 ; B = all ones
//   => D[m,n] = x(m) + x(m+16)
// Each lane sums its 8 D VGPRs (rows 0..7 on lanes 0-15, rows 8..15 on lanes
// 16-31), then one xor-16 shuffle-add yields the full wave sum in every lane.
__device__ __forceinline__ float wave_reduce_wmma(float x) {
  v2f a; a[0] = x;    a[1] = 0.0f;
  v2f b; b[0] = 1.0f; b[1] = 1.0f;
  v8f c = {};
  v8f d = __builtin_amdgcn_wmma_f32_16x16x4_f32(
      /*neg_a=*/false, a, /*neg_b=*/false, b,
      /*c_mod=*/(short)0, c, /*reuse_a=*/false, /*reuse_b=*/false);
  float s = ((d[0] + d[1]) + (d[2] + d[3])) + ((d[4] + d[5]) + (d[6] + d[7]));
  s += __shfl_xor(s, 16, 32);
  return s;
}

__global__ __launch_bounds__(BLOCK) void WeightsRegLoss_kernel(
    const float* __restrict__ weights,
    const int* __restrict__ px,
    const int* __restrict__ py,
    float* __restrict__ out) {
  const int x = *px;     // uniform scalar loads
  const int y = *py;
  const int tid = threadIdx.x;

  const float4* __restrict__ w4 = reinterpret_cast<const float4*>(weights);
  const int rowstride4 = IMG_W * CHUNKS;   // one image row in float4 units
  const int colstride4 = CHUNKS;           // one pixel in float4 units

  // Fixed (j, chunk) per thread -- division happens exactly once.
  const int j  = tid / CHUNKS;
  const int ch = tid - j * CHUNKS;

  float acc = 0.0f;
  if (tid < ROWTASKS) {
    long idx = (long)x * rowstride4 + (long)y * colstride4
             + (long)j * colstride4 + ch;
    // Rolling base: down(i) becomes base(i+1) -> 2 loads/iter instead of 3.
    float4 base = w4[idx];
    for (int i = 0; i < CELLS_1D; ++i) {
      const float4 dn = w4[idx + rowstride4];  // patch[i+1, j]
      const float4 rt = w4[idx + colstride4];  // patch[i,   j+1]

      const float dx0 = dn.x - base.x, dx1 = dn.y - base.y;
      const float dx2 = dn.z - base.z, dx3 = dn.w - base.w;
      const float dy0 = rt.x - base.x, dy1 = rt.y - base.y;
      const float dy2 = rt.z - base.z, dy3 = rt.w - base.w;
      acc = fmaf(dx0, dx0, acc); acc = fmaf(dx1, dx1, acc);
      acc = fmaf(dx2, dx2, acc); acc = fmaf(dx3, dx3, acc);
      acc = fmaf(dy0, dy0, acc); acc = fmaf(dy1, dy1, acc);
      acc = fmaf(dy2, dy2, acc); acc = fmaf(dy3, dy3, acc);

      base = dn;
      idx += rowstride4;
    }
  }
  // EXEC is full again here (structured control flow) -- safe for WMMA.

  // Stage 1: 32-lane reduction per wave via WMMA.
  const float wsum = wave_reduce_wmma(acc);

  __shared__ float partials[BLOCK / 32];
  const int wave = tid >> 5;
  const int lane = tid & 31;
  if (lane == 0) partials[wave] = wsum;
  __syncthreads();

  // Stage 2: wave 0 reduces the 32 wave partials, again via WMMA.
  if (wave == 0) {
    const float v = partials[lane];
    const float total = wave_reduce_wmma(v);
    if (lane == 0) out[0] = total * (1.0f / (float)NCH);
  }
}

extern "C" void kernel_launch(void* const* d_in, const int* in_sizes, int n_in,
                              void* d_out, int out_size, void* d_ws, size_t ws_size,
                              hipStream_t stream) {
  (void)in_sizes; (void)n_in; (void)out_size; (void)d_ws; (void)ws_size;
  const float* weights = (const float*)d_in[0];
  const int*   px      = (const int*)d_in[1];
  const int*   py      = (const int*)d_in[2];
  float*       out     = (float*)d_out;
  // Tiny latency-bound problem (~307 KB unique reads, ~0.28 MFLOP): a single
  // 1024-thread block (32 wave32s on one WGP) minimizes launch overhead while
  // the 32 waves hide global-load latency.
  WeightsRegLoss_kernel<<<1, BLOCK, 0, stream>>>(weights, px, py, out);
}